// MultiHeadedAttention_6794638263044
// MI455X (gfx1250) — compile-verified
//
#include <hip/hip_runtime.h>
#include <math.h>

// ---------------------------------------------------------------------------
// Quantum-inspired multi-head attention, collapsed algebraically:
//   * A = UC6^H diag(MZ6) UC6 is DIAGONAL (UC6 is a CNOT permutation)
//       => scores are a signed bilinear form of qd = |U0 psi|^2, kd = |U1 psi|^2
//   * vs = c c^H (rank-1)  => wvs = (Uw c)(Uw c)^H (rank-1)
//       => scan step is an 8-term gather-weighted update per rho entry
//   * kron3 unitaries applied as three 2x2 qubit sweeps
// Workload ~50 MFLOP / ~36KB traffic: pure latency problem on MI455X.
// 3 kernels: prep (1 WG) -> scan (32 WGs, 1 wave32 per (h,s) chain) ->
// epilogue GEMM z*W2^T via V_WMMA_F32_16X16X4_F32 (fp32 everywhere).
// ---------------------------------------------------------------------------

#define NH     8
#define DK     8
#define SEQL   32
#define DMODEL 64
#define PI_F   3.14159265358979323846f

struct cplx { float re, im; };
__device__ __forceinline__ cplx cmk(float r, float i){ cplx c; c.re=r; c.im=i; return c; }
__device__ __forceinline__ cplx cmul(cplx a, cplx b){
  return cmk(fmaf(a.re,b.re,-a.im*b.im), fmaf(a.re,b.im, a.im*b.re)); }
__device__ __forceinline__ cplx cmulc(cplx a, cplx b){ // a * conj(b)
  return cmk(fmaf(a.re,b.re, a.im*b.im), fmaf(a.im,b.re,-a.re*b.im)); }
__device__ __forceinline__ cplx cadd(cplx a, cplx b){ return cmk(a.re+b.re, a.im+b.im); }

// CNOT-ring permutation on 6 qubits (qubit 0 = MSB): gate k tests bit(5-k),
// flips bit(5-((k+1)%6)); forward applies k=0..5, inverse applies k=5..0.
__device__ __forceinline__ int ring_f(int b){
#pragma unroll
  for (int k = 0; k < 6; ++k) { if ((b >> (5 - k)) & 1) b ^= 1 << (5 - ((k + 1) % 6)); }
  return b;
}
__device__ __forceinline__ int ring_finv(int b){
#pragma unroll
  for (int k = 5; k >= 0; --k) { if ((b >> (5 - k)) & 1) b ^= 1 << (5 - ((k + 1) % 6)); }
  return b;
}
// a6[8a+b] = MZ6[f(8a+b)] = (-1)^popcount(f(8a+b))
__device__ __forceinline__ float sgn_ab(int a, int b){
  return (__popc(ring_f(a * 8 + b)) & 1) ? -1.0f : 1.0f;
}

// Apply kron3(u_q0,u_q1,u_q2) to an 8-vector via 3 qubit sweeps.
__device__ __forceinline__ void apply_sweep(cplx v[8], cplx u00, cplx u01, cplx u10, cplx u11, int st){
#pragma unroll
  for (int i = 0; i < 8; ++i) {
    if (i & st) continue;
    cplx va = v[i], vb = v[i + st];
    v[i]      = cadd(cmul(u00, va), cmul(u01, vb));
    v[i + st] = cadd(cmul(u10, va), cmul(u11, vb));
  }
}

// ------------------------- Kernel 1: prep (1 block) -------------------------
__global__ __launch_bounds__(256) void qattn_prep(
    const float* __restrict__ x, const float* __restrict__ pqc_w,
    const float* __restrict__ W1, const float* __restrict__ b1,
    float* __restrict__ qd, float* __restrict__ kd, float* __restrict__ cv)
{
  __shared__ float xq [SEQL][DK];
  __shared__ float psi[SEQL][DK];
  __shared__ cplx  uq [NH][3][3][4];      // per (h,g,qubit) 2x2 unitary

  const int t = threadIdx.x;

  // xq = x @ W1^T + b1   (32x64 * 64x8)
  {
    const int s = t >> 3, k = t & 7;
    float acc = b1[k];
#pragma unroll 8
    for (int d = 0; d < DMODEL; ++d) acc = fmaf(x[s * DMODEL + d], W1[k * DMODEL + d], acc);
    xq[s][k] = acc;
  }
  __syncthreads();
  // psi = row-normalize(xq)
  {
    const int s = t >> 3, k = t & 7;
    float nrm = 0.0f;
#pragma unroll
    for (int kk = 0; kk < DK; ++kk) { float v = xq[s][kk]; nrm = fmaf(v, v, nrm); }
    psi[s][k] = xq[s][k] * rsqrtf(nrm);
  }
  // per-(h,g,qubit) single-qubit unitaries u = rz(a2) ry(a1) rx(a0)
  if (t < NH * 3 * 3) {
    const int h = t / 9, g = (t / 3) % 3, q = t % 3;
    const float WMUL = 0.6324555320336759f;   // sqrt(2/5)
    const float a0 = pqc_w[(h * 3 + g) * 9 + q * 3 + 0] * WMUL;
    const float a1 = pqc_w[(h * 3 + g) * 9 + q * 3 + 1] * WMUL;
    const float a2 = pqc_w[(h * 3 + g) * 9 + q * 3 + 2] * WMUL;
    const float cx = cosf(0.5f * a0), sx = sinf(0.5f * a0);
    const float cy = cosf(0.5f * a1), sy = sinf(0.5f * a1);
    const float cz = cosf(0.5f * a2), sz = sinf(0.5f * a2);
    cplx m00 = cmk( cy * cx,  sy * sx), m01 = cmk(-sy * cx, -cy * sx);
    cplx m10 = cmk( sy * cx, -cy * sx), m11 = cmk( cy * cx, -sy * sx);
    cplx e0 = cmk(cz, -sz), e1 = cmk(cz, sz);
    uq[h][g][q][0] = cmul(e0, m00); uq[h][g][q][1] = cmul(e0, m01);
    uq[h][g][q][2] = cmul(e1, m10); uq[h][g][q][3] = cmul(e1, m11);
  }
  __syncthreads();
  // c[h,g,s] = kron3(u)*psi_s ; qd/kd = |c|^2 for g=0/1 ; cv = c for g=2
#pragma unroll
  for (int it = 0; it < 3; ++it) {
    const int task = t + 256 * it;                // 768 = 8*3*32 tasks
    const int h = task / 96, g = (task / 32) % 3, s = task & 31;
    cplx v[8];
#pragma unroll
    for (int a = 0; a < 8; ++a) v[a] = cmk(psi[s][a], 0.0f);
#pragma unroll
    for (int q = 0; q < 3; ++q)
      apply_sweep(v, uq[h][g][q][0], uq[h][g][q][1], uq[h][g][q][2], uq[h][g][q][3], 4 >> q);
    const int off = (h * SEQL + s) * 8;
    if (g == 0) {
#pragma unroll
      for (int a = 0; a < 8; ++a) qd[off + a] = fmaf(v[a].re, v[a].re, v[a].im * v[a].im);
    } else if (g == 1) {
#pragma unroll
      for (int a = 0; a < 8; ++a) kd[off + a] = fmaf(v[a].re, v[a].re, v[a].im * v[a].im);
    } else {
#pragma unroll
      for (int a = 0; a < 8; ++a) { cv[2 * (off + a)] = v[a].re; cv[2 * (off + a) + 1] = v[a].im; }
    }
  }
}

// ----------------- Kernel 2: scan (32 blocks, 1 wave per chain) -------------
__global__ __launch_bounds__(256) void qattn_scan(
    const float* __restrict__ qd, const float* __restrict__ kd,
    const float* __restrict__ cv, float* __restrict__ z)
{
  __shared__ cplx rho[8][64];                     // 8 chains/block, 8x8 complex
  const int wave = threadIdx.x >> 5, lane = threadIdx.x & 31;
  const int chain = blockIdx.x * 8 + wave;
  const int h = chain >> 5, s = chain & 31;

  // qsg[b] = sum_a sgn(a,b) * qd[h,s,a]
  float qsg[8];
  {
    float qdl[8];
#pragma unroll
    for (int a = 0; a < 8; ++a) qdl[a] = qd[(h * SEQL + s) * 8 + a];
#pragma unroll
    for (int b = 0; b < 8; ++b) {
      float acc = 0.0f;
#pragma unroll
      for (int a = 0; a < 8; ++a) acc = fmaf(sgn_ab(a, b), qdl[a], acc);
      qsg[b] = acc;
    }
  }
  // lane j holds key-diag / value-vector for step j (broadcast via shuffles)
  float kdl[8]; cplx cvl[8];
#pragma unroll
  for (int b = 0; b < 8; ++b) {
    kdl[b] = kd[(h * SEQL + lane) * 8 + b];
    cvl[b] = cmk(cv[2 * ((h * SEQL + lane) * 8 + b)], cv[2 * ((h * SEQL + lane) * 8 + b) + 1]);
  }
  // permutation-gather indices for this lane's two rho entries
  int p1[2][8], p2[2][8];
#pragma unroll
  for (int tt = 0; tt < 2; ++tt) {
    const int e = lane + 32 * tt, a = e >> 3, c = e & 7;
#pragma unroll
    for (int b = 0; b < 8; ++b) { p1[tt][b] = ring_finv(a * 8 + b); p2[tt][b] = ring_finv(c * 8 + b); }
  }

  for (int j = 0; j < SEQL; ++j) {
    // phi = raw * pi ;  raw = sum_b qsg[b] * kd[h,j,b]
    float raw = 0.0f;
#pragma unroll
    for (int b = 0; b < 8; ++b) raw = fmaf(qsg[b], __shfl(kdl[b], j, 32), raw);
    const float phi = raw * PI_F;
    // u1 = rz(phi) ry(phi/2) rx(phi/2) (rx,ry share the half-angle phi/4)
    const float ch = cosf(0.25f * phi), sh = sinf(0.25f * phi);
    const float cz = cosf(0.50f * phi), sz = sinf(0.50f * phi);
    cplx m00 = cmk( ch * ch,  sh * sh), m01 = cmk(-sh * ch, -ch * sh);
    cplx m10 = cmk( sh * ch, -ch * sh), m11 = cmk( ch * ch, -sh * sh);
    cplx e0 = cmk(cz, -sz), e1 = cmk(cz, sz);
    cplx u00 = cmul(e0, m00), u01 = cmul(e0, m01), u10 = cmul(e1, m10), u11 = cmul(e1, m11);
    // d = kron3(u1,u1,u1) * c_v[h,j]
    cplx d[8];
#pragma unroll
    for (int b = 0; b < 8; ++b) d[b] = cmk(__shfl(cvl[b].re, j, 32), __shfl(cvl[b].im, j, 32));
#pragma unroll
    for (int q = 0; q < 3; ++q) apply_sweep(d, u00, u01, u10, u11, 4 >> q);

    if (j == 0) {                                  // rho <- d d^H
#pragma unroll
      for (int tt = 0; tt < 2; ++tt) {
        const int e = lane + 32 * tt;
        rho[wave][e] = cmulc(d[e >> 3], d[e & 7]);
      }
      __syncthreads();
    } else {                                       // permuted rank-1 partial trace
      cplx nv[2];
#pragma unroll
      for (int tt = 0; tt < 2; ++tt) {
        cplx acc = cmk(0.0f, 0.0f);
#pragma unroll
        for (int b = 0; b < 8; ++b) {
          const int i1 = p1[tt][b], i2 = p2[tt][b];
          acc = cadd(acc, cmul(rho[wave][(i1 >> 3) * 8 + (i2 >> 3)], cmulc(d[i1 & 7], d[i2 & 7])));
        }
        nv[tt] = acc;
      }
      __syncthreads();
#pragma unroll
      for (int tt = 0; tt < 2; ++tt) rho[wave][lane + 32 * tt] = nv[tt];
      __syncthreads();
    }
  }
  if (lane < 8) z[s * 64 + h * 8 + lane] = rho[wave][lane * 9].re;  // diag, real
}

// ------------- Kernel 3: out = z @ W2^T + b2 via WMMA f32 16x16x4 -----------
typedef float v2f __attribute__((ext_vector_type(2)));
typedef float v8f __attribute__((ext_vector_type(8)));

__global__ __launch_bounds__(256) void qattn_out(
    const float* __restrict__ z, const float* __restrict__ W2,
    const float* __restrict__ b2, float* __restrict__ out)
{
  const int wave = threadIdx.x >> 5, lane = threadIdx.x & 31;
  const int mi = wave >> 2, ni = wave & 3;        // 2x4 tiles of 16x16, 8 waves
  const int col = lane & 15;                      // N (B/C/D) and M-row (A)
  const int hi  = lane >> 4;                      // K-half for A/B, M-half for C/D

  v8f acc;
  const float bias = b2[ni * 16 + col];
#pragma unroll
  for (int r = 0; r < 8; ++r) acc[r] = bias;

#pragma unroll
  for (int k0 = 0; k0 < DMODEL; k0 += 4) {
    const int kb = k0 + (hi << 1);
    v2f a, b;
    // A 16x4: lanes 0-15 hold K=0,1 ; lanes 16-31 hold K=2,3 (ISA 7.12.2)
    a.x = z[(mi * 16 + col) * DMODEL + kb];
    a.y = z[(mi * 16 + col) * DMODEL + kb + 1];
    // B 4x16 (= W2^T tile): B[k][n] = W2[n][k]
    b.x = W2[(ni * 16 + col) * DMODEL + kb];
    b.y = W2[(ni * 16 + col) * DMODEL + kb + 1];
    acc = __builtin_amdgcn_wmma_f32_16x16x4_f32(false, a, false, b, (short)0, acc, false, false);
  }
#pragma unroll
  for (int r = 0; r < 8; ++r) {
    const int M = r + (hi << 3);                  // C/D: VGPR r -> M=r / M=r+8
    out[(mi * 16 + M) * DMODEL + ni * 16 + col] = acc[r];
  }
}

// ---------------------------------------------------------------------------
extern "C" void kernel_launch(void* const* d_in, const int* in_sizes, int n_in,
                              void* d_out, int out_size, void* d_ws, size_t ws_size,
                              hipStream_t stream) {
  (void)in_sizes; (void)n_in; (void)out_size; (void)ws_size;
  const float* x     = (const float*)d_in[0];
  const float* pqc_w = (const float*)d_in[1];
  const float* W1    = (const float*)d_in[2];
  const float* b1    = (const float*)d_in[3];
  const float* W2    = (const float*)d_in[4];
  const float* b2    = (const float*)d_in[5];

  float* ws = (float*)d_ws;
  float* qd = ws;           // H*SEQ*8           = 2048 f32
  float* kd = ws + 2048;    // H*SEQ*8           = 2048 f32
  float* cv = ws + 4096;    // H*SEQ*8 complex   = 4096 f32
  float* z  = ws + 8192;    // SEQ*64            = 2048 f32

  qattn_prep<<<1, 256, 0, stream>>>(x, pqc_w, W1, b1, qd, kd, cv);
  qattn_scan<<<32, 256, 0, stream>>>(qd, kd, cv, z);
  qattn_out <<<1, 256, 0, stream>>>(z, W2, b2, (float*)d_out);
}